// LuongAttention_10977936408725
// MI455X (gfx1250) — compile-verified
//
#include <hip/hip_runtime.h>
#include <math.h>

typedef __attribute__((ext_vector_type(2))) float v2f;
typedef __attribute__((ext_vector_type(8))) float v8f;
typedef __attribute__((ext_vector_type(4))) unsigned int v4u;
typedef __attribute__((ext_vector_type(8))) int v8i;
typedef __attribute__((ext_vector_type(4))) int v4i;

#define B_DIM 32
#define S_DIM 8192
#define D_DIM 512
#define WG_PER_BATCH 32        // S-chunks per batch
#define ROWS_PER_WG 256        // S_DIM / WG_PER_BATCH
#define ROWS_PER_WAVE 64       // each wave streams 64 rows (4 tiles of 16)
#define ROWS_PER_TILE 16
#define ROUNDS 4               // ROWS_PER_WAVE / ROWS_PER_TILE
#define LDS_STRIDE 516         // 512 + 4 pad floats: 16B align, banks spread by 4
#define TILE_FLOATS (ROWS_PER_TILE * LDS_STRIDE)   // 8256 floats = 33024 B
#define PARTIALS_PER_BATCH 128 // WG_PER_BATCH * 4 waves

// ---------------------------------------------------------------------------
// TDM: DMA one 16x512 fp32 tile (row-major, global) into LDS with row stride
// 516 floats, using iterate mode: 16 iterations of one 512-float row,
// lds_addr_increment = 516, global_addr_increment = 512 (data_size units).
// D# bitfields per CDNA5 ISA ch.8. 6-arg builtin on this toolchain:
// (v4u g0, v8i g1, v4i g2, v4i g3, v8i zero, i32 cpol).
// ---------------------------------------------------------------------------
__device__ __forceinline__ void tdm_load_tile16(const float* gsrc, float* ldst) {
    unsigned long long ga = (unsigned long long)(uintptr_t)gsrc;
    unsigned int lds_off  = (unsigned int)(uintptr_t)ldst;  // low 32b = LDS byte offset

    v4u g0;
    g0[0] = 1u;                                   // count=1, user descriptor
    g0[1] = lds_off;                              // lds_addr (bytes)
    g0[2] = (unsigned int)(ga & 0xFFFFFFFFu);     // global_addr[31:0]
    g0[3] = (unsigned int)((ga >> 32) & 0x01FFFFFFu) | (2u << 30);  // addr[56:32] | type=2

    v8i g1;
    g1[0] = (int)((2u << 16) | (1u << 19));       // data_size=4B, iterate_enable=1
    g1[1] = (int)(512u << 16);                    // tensor_dim0[15:0]=512 (barrier addr=0)
    g1[2] = (int)(1u << 16);                      // tensor_dim0 hi=0 | tensor_dim1=1
    g1[3] = (int)(512u << 16);                    // tensor_dim1 hi=0 | tile_dim0=512
    g1[4] = 1;                                    // tile_dim1=1, tile_dim2=0
    g1[5] = 512;                                  // tensor_dim0_stride lo (elements)
    g1[6] = 0;                                    // stride0 hi | tensor_dim1_stride lo
    g1[7] = 0;                                    // tensor_dim1_stride hi

    v4i g2;
    g2[0] = 0;                                    // tensor_dim2 (unused)
    g2[1] = LDS_STRIDE;                           // lds_addr_increment (elements)
    g2[2] = 512;                                  // global_addr_increment lo (elements)
    g2[3] = (int)(15u << 16);                     // gai hi=0 | iterate_count=15 -> 16 rows

    v4i g3 = {0, 0, 0, 0};
    v8i g4 = {0, 0, 0, 0, 0, 0, 0, 0};            // unused trailing group (zero)

    __builtin_amdgcn_tensor_load_to_lds(g0, g1, g2, g3, g4, 0);
}

// ---------------------------------------------------------------------------
// Per-tile compute: 16 row-scores via V_WMMA_F32_16X16X4_F32, online-softmax
// update of (m, l), context accumulation into 16 per-lane fp32 accumulators.
// Both A and B fragments come straight from LDS with identical addressing:
//   A: enc[row=r, k+2*hf .. +1]  (banks 4r+2hf, conflict-free)
//   B: dec[k+2*hf .. +1]         (same address within a half -> LDS broadcast)
// ---------------------------------------------------------------------------
__device__ __forceinline__ void compute_tile(const float* __restrict__ wcur,
                                             const float* __restrict__ dlds,
                                             int hf, int r, int lane,
                                             float& m_run, float& l_run,
                                             float (&acc)[16]) {
    const float* arow = wcur + r * LDS_STRIDE + 2 * hf;
    const float* brow = dlds + 2 * hf;
    v8f cacc = {0.f, 0.f, 0.f, 0.f, 0.f, 0.f, 0.f, 0.f};
    #pragma unroll 8
    for (int k = 0; k < D_DIM; k += 4) {
        v2f af = *(const v2f*)(arow + k);
        v2f bf = *(const v2f*)(brow + k);
        cacc = __builtin_amdgcn_wmma_f32_16x16x4_f32(
            false, af, false, bf, (short)0, cacc, false, false);
    }

    // D layout: VGPR j holds the score of row M = j + 8*hf (all N equal).
    float sc[16];
    #pragma unroll
    for (int j = 0; j < 8; ++j) {
        float mine = cacc[j];
        float oth  = __shfl_xor(mine, 16);
        sc[j]     = hf ? oth : mine;   // rows 0..7
        sc[j + 8] = hf ? mine : oth;   // rows 8..15
    }

    // ---- online softmax ----
    float tmax = sc[0];
    #pragma unroll
    for (int j = 1; j < 16; ++j) tmax = fmaxf(tmax, sc[j]);
    float m_new = fmaxf(m_run, tmax);
    float corr  = __expf(m_run - m_new);
    l_run *= corr;
    #pragma unroll
    for (int i = 0; i < 16; ++i) acc[i] *= corr;
    float p[16];
    #pragma unroll
    for (int j = 0; j < 16; ++j) {
        p[j] = __expf(sc[j] - m_new);
        l_run += p[j];
    }
    m_run = m_new;

    // ---- phase B: context accumulation, lane owns d = blk*128 + lane*4 + i
    // (ds_load_b128, banks 4*lane, conflict-free per half-wave).
    const float* bbase = wcur + lane * 4;
    #pragma unroll
    for (int j = 0; j < 16; ++j) {
        const float* rp = bbase + j * LDS_STRIDE;
        float pj = p[j];
        #pragma unroll
        for (int blk = 0; blk < 4; ++blk) {
            float4 e = *(const float4*)(rp + blk * 128);
            acc[blk * 4 + 0] = fmaf(pj, e.x, acc[blk * 4 + 0]);
            acc[blk * 4 + 1] = fmaf(pj, e.y, acc[blk * 4 + 1]);
            acc[blk * 4 + 2] = fmaf(pj, e.z, acc[blk * 4 + 2]);
            acc[blk * 4 + 3] = fmaf(pj, e.w, acc[blk * 4 + 3]);
        }
    }
}

// ---------------------------------------------------------------------------
// Kernel 1: per-wave streaming online-softmax partials, TDM double-buffered.
// Each of 4 waves independently owns 64 rows: DMA tile n+1 into its spare LDS
// buffer while computing tile n; sync via s_wait_tensorcnt only (one block
// barrier at start after staging the decoder row into LDS).
// encoder_hiddens is read from HBM exactly once (~22 us roofline floor).
// ---------------------------------------------------------------------------
extern "C" __global__ __launch_bounds__(128)
void attn_partial_kernel(const float* __restrict__ dec,
                         const float* __restrict__ enc,
                         float* __restrict__ accbuf,
                         float* __restrict__ mbuf,
                         float* __restrict__ lbuf) {
    extern __shared__ float tile[];   // 4 waves * 2 * TILE_FLOATS + 512 (dec) = 266240 B

    const int b    = blockIdx.y;
    const int cg   = blockIdx.x;
    const int tid  = threadIdx.x;
    const int lane = tid & 31;
    const int wave = tid >> 5;
    const int hf   = lane >> 4;
    const int r    = lane & 15;

    const float* gwave = enc + ((size_t)b * S_DIM
                              + (size_t)cg * ROWS_PER_WG
                              + (size_t)wave * ROWS_PER_WAVE) * D_DIM;

    float* wbuf0 = tile + wave * (2 * TILE_FLOATS);
    float* wbuf1 = wbuf0 + TILE_FLOATS;
    float* dlds  = tile + 4 * (2 * TILE_FLOATS);   // 512-float decoder row

    // prime the pipeline: tile 0 -> buf0 (TDM runs while we stage dec)
    tdm_load_tile16(gwave, wbuf0);

    // stage decoder row (2 KB) into LDS: 128 threads x float4, coalesced
    {
        float4 v = *(const float4*)(dec + b * D_DIM + tid * 4);
        *(float4*)(dlds + tid * 4) = v;
    }
    __syncthreads();

    float m_run = -INFINITY;
    float l_run = 0.0f;
    float acc[16];
    #pragma unroll
    for (int i = 0; i < 16; ++i) acc[i] = 0.0f;

    #pragma unroll 1
    for (int round = 0; round < ROUNDS - 1; ++round) {
        float* cur = (round & 1) ? wbuf1 : wbuf0;
        float* nxt = (round & 1) ? wbuf0 : wbuf1;
        // prefetch next tile while computing current
        tdm_load_tile16(gwave + (size_t)(round + 1) * ROWS_PER_TILE * D_DIM, nxt);
        __builtin_amdgcn_s_wait_tensorcnt(1);   // current tile landed
        compute_tile(cur, dlds, hf, r, lane, m_run, l_run, acc);
    }
    __builtin_amdgcn_s_wait_tensorcnt(0);       // last tile landed
    compute_tile(((ROUNDS - 1) & 1) ? wbuf1 : wbuf0,
                 dlds, hf, r, lane, m_run, l_run, acc);

    // ---- write this wave's partial (m, l, acc[512]) ----
    const int pidx = b * PARTIALS_PER_BATCH + cg * 4 + wave;
    float* ap = accbuf + (size_t)pidx * D_DIM;
    #pragma unroll
    for (int blk = 0; blk < 4; ++blk) {
        float4 o;
        o.x = acc[blk * 4 + 0];
        o.y = acc[blk * 4 + 1];
        o.z = acc[blk * 4 + 2];
        o.w = acc[blk * 4 + 3];
        *(float4*)(ap + blk * 128 + lane * 4) = o;
    }
    if (lane == 0) {
        mbuf[pidx] = m_run;
        lbuf[pidx] = l_run;
    }
}

// ---------------------------------------------------------------------------
// Kernel 2: combine 128 partials per batch (flash-style rescale + merge).
// ~8 MB traffic, negligible vs the 512 MB stream.
// ---------------------------------------------------------------------------
extern "C" __global__ __launch_bounds__(128)
void attn_combine_kernel(const float* __restrict__ accbuf,
                         const float* __restrict__ mbuf,
                         const float* __restrict__ lbuf,
                         float* __restrict__ out) {
    __shared__ float wsh[PARTIALS_PER_BATCH];
    __shared__ float red[PARTIALS_PER_BATCH];

    const int b = blockIdx.x;
    const int t = threadIdx.x;   // 0..127

    float mg = mbuf[b * PARTIALS_PER_BATCH + t];
    float lg = lbuf[b * PARTIALS_PER_BATCH + t];

    red[t] = mg;
    __syncthreads();
    for (int off = 64; off > 0; off >>= 1) {
        if (t < off) red[t] = fmaxf(red[t], red[t + off]);
        __syncthreads();
    }
    float M = red[0];
    __syncthreads();

    float wg = __expf(mg - M);
    wsh[t] = wg;
    red[t] = wg * lg;
    __syncthreads();
    for (int off = 64; off > 0; off >>= 1) {
        if (t < off) red[t] += red[t + off];
        __syncthreads();
    }
    float invL = 1.0f / red[0];
    __syncthreads();

    const float* ab = accbuf + (size_t)b * PARTIALS_PER_BATCH * D_DIM;
    #pragma unroll
    for (int dblk = 0; dblk < 4; ++dblk) {
        int d = dblk * 128 + t;
        float ctx = 0.0f;
        for (int g = 0; g < PARTIALS_PER_BATCH; ++g)
            ctx = fmaf(wsh[g], ab[(size_t)g * D_DIM + d], ctx);
        out[b * D_DIM + d] = ctx * invL;
    }
}

// ---------------------------------------------------------------------------
extern "C" void kernel_launch(void* const* d_in, const int* in_sizes, int n_in,
                              void* d_out, int out_size, void* d_ws, size_t ws_size,
                              hipStream_t stream) {
    const float* dec = (const float*)d_in[0];   // (32, 512) fp32
    const float* enc = (const float*)d_in[1];   // (32, 8192, 512) fp32
    float* out = (float*)d_out;                 // (32, 512) fp32

    float* accbuf = (float*)d_ws;                                        // 8 MB
    float* mbuf   = accbuf + (size_t)B_DIM * PARTIALS_PER_BATCH * D_DIM;
    float* lbuf   = mbuf + B_DIM * PARTIALS_PER_BATCH;

    dim3 grid1(WG_PER_BATCH, B_DIM);
    size_t shmem = ((size_t)4 * 2 * TILE_FLOATS + D_DIM) * sizeof(float); // 266240 B
    attn_partial_kernel<<<grid1, 128, shmem, stream>>>(dec, enc, accbuf, mbuf, lbuf);
    attn_combine_kernel<<<B_DIM, 128, 0, stream>>>(accbuf, mbuf, lbuf, out);
}